// Physics2D_25099788878059
// MI455X (gfx1250) — compile-verified
//
#include <hip/hip_runtime.h>

typedef float v2f __attribute__((ext_vector_type(2)));
typedef float v8f __attribute__((ext_vector_type(8)));

namespace {
constexpr int BB = 4, SS = 2048, NN = BB * SS, KNEG = 4;
constexpr float SOFT = 1e-6f, EPSV = 1e-9f, MARGINF = 1.0f, KEF = 1e-3f;
constexpr float CUT_INV = 10.0f; // 1/CUTOFF
}

__device__ __forceinline__ v8f wmma_k4(v2f a, v2f b, v8f c) {
  // D = A(16x4) x B(4x16) + C, fp32 — V_WMMA_F32_16X16X4_F32
  return __builtin_amdgcn_wmma_f32_16x16x4_f32(false, a, false, b, (short)0, c,
                                               false, false);
}

__device__ __forceinline__ float rcp_fast(float x) {
  return __builtin_amdgcn_rcpf(x);      // v_rcp_f32 (~1 ulp)
}
__device__ __forceinline__ float sqrt_fast(float x) {
  return __builtin_amdgcn_sqrtf(x);     // v_sqrt_f32
}

__device__ __forceinline__ float wave_sum32(float x) {
#pragma unroll
  for (int off = 16; off > 0; off >>= 1) x += __shfl_xor(x, off, 32);
  return x;
}

// sum across a 16-lane half (xor offsets < 16 never cross halves)
__device__ __forceinline__ float half_sum16(float x) {
#pragma unroll
  for (int off = 8; off > 0; off >>= 1) x += __shfl_xor(x, off, 32);
  return x;
}

// ---------------------------------------------------------------------------
// forces: one wave per 16-target tile.
//   A (fixed)  = target tile positions, 16x4 (K=2,3 zero-padded)
//   B (stream) = source tile positions, 4x16
//   D[i,j] = p_i . p_j ; lane holds source col j = lane&15 (its OWN loaded
//   point -> shuffle-free inner loop), target rows i = v + 8*(lane>=16).
// Per-row sums reduced across the 16-lane half once at the end.
// ---------------------------------------------------------------------------
__global__ __launch_bounds__(256) void forces_kernel(
    const float* __restrict__ pos,   // [B,S,2]
    const float* __restrict__ mass,  // [B,S]
    const float* __restrict__ Gptr, const float* __restrict__ repptr,
    float* __restrict__ out)         // [B,S,2]
{
  const int lane = threadIdx.x & 31;
  const int waveId = blockIdx.x * (blockDim.x >> 5) + (threadIdx.x >> 5);
  const int tilesPerBatch = SS / 16;                 // 128
  const int b = waveId / tilesPerBatch;
  const int i0 = (waveId % tilesPerBatch) * 16;
  if (b >= BB) return;                               // wave-uniform

  const int col = lane & 15;
  const bool hi = lane >= 16;
  const float Gp  = fmaxf(Gptr[0], 0.0f);
  const float rep = fmaxf(repptr[0], 0.0f);

  // ---- fixed A matrix: targets (lanes<16 give K=0,1; lanes>=16 give 0) ----
  const float2 pi2 = ((const float2*)pos)[b * SS + i0 + col];
  const float mi_own = mass[b * SS + i0 + col];
  v2f amat; amat.x = hi ? 0.f : pi2.x; amat.y = hi ? 0.f : pi2.y;

  // one-time gather of per-row target scalars (rows v + 8*hi)
  float xi[8], yi[8], mi[8], r2i[8];
#pragma unroll
  for (int v = 0; v < 8; ++v) {
    const int src = v + (hi ? 8 : 0);
    xi[v] = __shfl(pi2.x, src, 32);
    yi[v] = __shfl(pi2.y, src, 32);
    mi[v] = __shfl(mi_own, src, 32);
    r2i[v] = xi[v] * xi[v] + yi[v] * yi[v];
  }
  const int rowbase = i0 + (hi ? 8 : 0);

  float accW[8], accWx[8], accWy[8];
#pragma unroll
  for (int v = 0; v < 8; ++v) { accW[v] = 0.f; accWx[v] = 0.f; accWy[v] = 0.f; }

  for (int j0 = 0; j0 < SS; j0 += 16) {
    const float2 pj2 = ((const float2*)pos)[b * SS + j0 + col];
    const float mj = mass[b * SS + j0 + col];
    const float r2j = pj2.x * pj2.x + pj2.y * pj2.y;
    v2f bmat; bmat.x = hi ? 0.f : pj2.x; bmat.y = hi ? 0.f : pj2.y;

    v8f c = {0.f, 0.f, 0.f, 0.f, 0.f, 0.f, 0.f, 0.f};
    const v8f d = wmma_k4(amat, bmat, c);  // d[v] = p_{i0+v+8hi} . p_{j0+col}
    const int jg = j0 + col;

#pragma unroll
    for (int v = 0; v < 8; ++v) {
      const float d2raw = fmaxf(r2i[v] + r2j - 2.0f * d[v], 0.0f) + SOFT;
      const bool diag = (rowbase + v) == jg;
      const float ds2 = diag ? 1.0f : d2raw;          // dist_sq_safe
      const float dist = sqrt_fast(ds2);              // dist_safe (==1 diag)
      const float att = Gp * (mi[v] * mj) * rcp_fast(ds2);
      const float lin = fmaxf(1.0f - dist * CUT_INV, 0.0f);
      const float repl = rep * lin * rcp_fast(ds2 * dist + EPSV);
      const float w = (repl - att) * rcp_fast(dist + EPSV);
      accW[v]  += w;
      accWx[v] += w * pj2.x;
      accWy[v] += w * pj2.y;
    }
  }

  // reduce each row's partial sums across its 16-lane half
#pragma unroll
  for (int v = 0; v < 8; ++v) {
    accW[v]  = half_sum16(accW[v]);
    accWx[v] = half_sum16(accWx[v]);
    accWy[v] = half_sum16(accWy[v]);
  }
  // lanes 0-7 write rows 0-7; lanes 16-23 write rows 8-15
  float fw = 0.f, fwx = 0.f, fwy = 0.f, xs = 0.f, ys = 0.f;
#pragma unroll
  for (int v = 0; v < 8; ++v) {
    if ((lane & 7) == v) {
      fw = accW[v]; fwx = accWx[v]; fwy = accWy[v]; xs = xi[v]; ys = yi[v];
    }
  }
  if ((lane & 15) < 8) {
    float2 f;
    f.x = fwx - xs * fw;
    f.y = fwy - ys * fw;
    ((float2*)out)[b * SS + rowbase + (lane & 7)] = f;
  }
}

// ---------------------------------------------------------------------------
// well kernel: sum over rows of min_w ||row - well_w||^2  (16 wells, D=8)
// A = wells (16xK), B = 16-row point tile; K=8 via two C-chained K=4 WMMAs.
// ---------------------------------------------------------------------------
__global__ __launch_bounds__(256) void well_kernel(
    const float* __restrict__ rows, int nrows,
    const float* __restrict__ wells,   // [16,8]
    float* __restrict__ accum)         // single float
{
  const int lane = threadIdx.x & 31;
  const int col = lane & 15;
  const bool hi = lane >= 16;

  // loop-invariant: well row `col`, dims selected per half-wave for A layout
  const float4 wlo = ((const float4*)wells)[col * 2 + 0];
  const float4 whi = ((const float4*)wells)[col * 2 + 1];
  const float r2w_own = wlo.x * wlo.x + wlo.y * wlo.y + wlo.z * wlo.z +
                        wlo.w * wlo.w + whi.x * whi.x + whi.y * whi.y +
                        whi.z * whi.z + whi.w * whi.w;
  v2f a1, a2;
  a1.x = hi ? wlo.z : wlo.x;  a1.y = hi ? wlo.w : wlo.y;   // dims 0..3
  a2.x = hi ? whi.z : whi.x;  a2.y = hi ? whi.w : whi.y;   // dims 4..7
  float r2w[8];
#pragma unroll
  for (int v = 0; v < 8; ++v) r2w[v] = __shfl(r2w_own, v + (hi ? 8 : 0), 32);

  const int wavesTotal = gridDim.x * (blockDim.x >> 5);
  const int waveId = blockIdx.x * (blockDim.x >> 5) + (threadIdx.x >> 5);
  const int ntiles = nrows / 16;     // all callers are multiples of 16

  float acc = 0.f;
  for (int t = waveId; t < ntiles; t += wavesTotal) {
    const int r = t * 16 + col;
    const float4 plo = ((const float4*)rows)[r * 2 + 0];
    const float4 phi = ((const float4*)rows)[r * 2 + 1];
    const float r2p = plo.x * plo.x + plo.y * plo.y + plo.z * plo.z +
                      plo.w * plo.w + phi.x * phi.x + phi.y * phi.y +
                      phi.z * phi.z + phi.w * phi.w;
    v2f b1, b2;
    b1.x = hi ? plo.z : plo.x;  b1.y = hi ? plo.w : plo.y;
    b2.x = hi ? phi.z : phi.x;  b2.y = hi ? phi.w : phi.y;

    v8f c = {0.f, 0.f, 0.f, 0.f, 0.f, 0.f, 0.f, 0.f};
    v8f d = wmma_k4(a1, b1, c);
    d = wmma_k4(a2, b2, d);          // d[v] = well_{v+8*hi} . row_{t*16+col}

    float mn = 3.4e38f;
#pragma unroll
    for (int v = 0; v < 8; ++v) {
      const float d2 = fmaxf(r2p + r2w[v] - 2.0f * d[v], 0.0f);
      mn = fminf(mn, d2);
    }
    mn = fminf(mn, __shfl_xor(mn, 16, 32));   // min over all 16 wells
    if (!hi) acc += mn;                       // count each row once
  }
  acc = wave_sum32(acc);
  if (lane == 0) atomicAdd(accum, acc);
}

// ---------------------------------------------------------------------------
// elementwise reductions for the remaining loss terms
// ---------------------------------------------------------------------------
__global__ __launch_bounds__(256) void reduce_pos_kernel(
    const float* __restrict__ pos2, const float* __restrict__ pp2,
    const float* __restrict__ mass, const float* __restrict__ pmass,
    const float* __restrict__ vel, float* __restrict__ a)
{
  float spp = 0.f, sm = 0.f, ske = 0.f;
  for (int i = blockIdx.x * blockDim.x + threadIdx.x; i < NN;
       i += gridDim.x * blockDim.x) {
    const float2 p = ((const float2*)pos2)[i];
    const float2 q = ((const float2*)pp2)[i];
    const float dx = p.x - q.x, dy = p.y - q.y;
    const float dist = sqrt_fast(dx * dx + dy * dy) + EPSV;
    spp += mass[i] * pmass[i] * rcp_fast(dist);
    sm += mass[i];
    const float2 v = ((const float2*)vel)[i];
    ske += v.x * v.x + v.y * v.y;
  }
  spp = wave_sum32(spp); sm = wave_sum32(sm); ske = wave_sum32(ske);
  if ((threadIdx.x & 31) == 0) {
    atomicAdd(&a[3], spp);
    atomicAdd(&a[5], sm);
    atomicAdd(&a[6], ske);
  }
}

__global__ __launch_bounds__(256) void reduce_neg_kernel(
    const float* __restrict__ pos2, const float* __restrict__ np2,
    const float* __restrict__ emass, const float* __restrict__ nmass,
    float* __restrict__ a)
{
  float spn = 0.f;
  for (int k = blockIdx.x * blockDim.x + threadIdx.x; k < NN * KNEG;
       k += gridDim.x * blockDim.x) {
    const int i = k >> 2;   // jnp.repeat(pos, 4, axis=0)
    const float2 p = ((const float2*)pos2)[i];
    const float2 q = ((const float2*)np2)[k];
    const float dx = p.x - q.x, dy = p.y - q.y;
    const float dist = sqrt_fast(dx * dx + dy * dy) + EPSV;
    spn += emass[k] * nmass[k] * rcp_fast(dist);
  }
  spn = wave_sum32(spn);
  if ((threadIdx.x & 31) == 0) atomicAdd(&a[4], spn);
}

__global__ void init_ws_kernel(float* a) {
  if (threadIdx.x < 8) a[threadIdx.x] = 0.f;
}

// Entropy term cancels exactly: F_pos - F_neg == U_pos - U_neg.
// mean over repeat(x,4) == mean over x, so S1 is reused with weight 4.
__global__ void finalize_kernel(const float* __restrict__ a,
                                const float* __restrict__ Gptr,
                                const float* __restrict__ wsptr,
                                float* __restrict__ out_loss) {
  const float Gp = fmaxf(Gptr[0], 0.f);
  const float ws = fmaxf(wsptr[0], 0.f);
  const float S1 = a[0], S2 = a[1], S3 = a[2];
  const float Spp = a[3], Spn = a[4], Sm = a[5], Ske = a[6];
  const float fN = (float)NN;
  const float U_pos = (-Gp) * Spp / fN + ws * (S1 + S2) / (2.f * fN);
  const float U_neg = (-Gp) * Spn / (4.f * fN) + ws * (4.f * S1 + S3) / (8.f * fN);
  const float fe = fmaxf(MARGINF + U_pos - U_neg, 0.f);
  const float ke = KEF * 0.5f * (Sm / fN) * (Ske / fN);
  out_loss[0] = fe + ke;
}

extern "C" void kernel_launch(void* const* d_in, const int* in_sizes, int n_in,
                              void* d_out, int out_size, void* d_ws, size_t ws_size,
                              hipStream_t stream) {
  const float* pos2d  = (const float*)d_in[0];   // [4,2048,2]
  const float* masses = (const float*)d_in[1];   // [4,2048,1]
  const float* vel    = (const float*)d_in[2];   // [8192,2]
  const float* pos8   = (const float*)d_in[3];   // [8192,8]
  const float* pp2    = (const float*)d_in[4];   // [8192,2]
  const float* pp8    = (const float*)d_in[5];   // [8192,8]
  const float* pmass  = (const float*)d_in[6];   // [8192,1]
  const float* np2    = (const float*)d_in[7];   // [32768,2]
  const float* np8    = (const float*)d_in[8];   // [32768,8]
  const float* nmass  = (const float*)d_in[9];   // [32768,1]
  const float* emass  = (const float*)d_in[10];  // [32768,1]
  const float* G      = (const float*)d_in[11];
  const float* rep    = (const float*)d_in[12];
  const float* wells  = (const float*)d_in[13];  // [16,8]
  const float* wstr   = (const float*)d_in[14];
  (void)in_sizes; (void)n_in; (void)out_size; (void)ws_size;

  float* out = (float*)d_out;          // forces [16384] then loss [1]
  float* acc = (float*)d_ws;           // 8 scratch floats

  init_ws_kernel<<<1, 32, 0, stream>>>(acc);
  // 4*2048/16 = 512 waves; 8 waves/block -> 64 blocks
  forces_kernel<<<64, 256, 0, stream>>>(pos2d, masses, G, rep, out);
  well_kernel<<<64, 256, 0, stream>>>(pos8, NN, wells, acc + 0);
  well_kernel<<<64, 256, 0, stream>>>(pp8, NN, wells, acc + 1);
  well_kernel<<<128, 256, 0, stream>>>(np8, NN * KNEG, wells, acc + 2);
  reduce_pos_kernel<<<32, 256, 0, stream>>>(pos2d, pp2, masses, pmass, vel, acc);
  reduce_neg_kernel<<<128, 256, 0, stream>>>(pos2d, np2, emass, nmass, acc);
  finalize_kernel<<<1, 1, 0, stream>>>(acc, G, wstr, out + NN * 2);
}